// UpFirDn2D_80281528697498
// MI455X (gfx1250) — compile-verified
//
#include <hip/hip_runtime.h>
#include <hip/hip_bf16.h>
#include <stdint.h>

// UpFirDn2D: up=2, down=1, K=4, pad=(2,1). x:(8,64,256,256) f32 -> (8,64,512,512) f32.
// Memory-bound (~0.6 GB traffic, ~1 GFLOP -> ~1.6 FLOP/B; WMMA irrelevant).
// CDNA5 strategy: async global->LDS halo staging (ASYNCcnt), wave32-coalesced
// b128 non-temporal output stores, 2x(2x2) outputs per thread.

#define H_IN    256
#define W_IN    256
#define W_OUT   512
#define TILE_W  32                   // input cols per block
#define TILE_H  16                   // input rows per block
#define HALO_W  (TILE_W + 2)         // 34
#define HALO_H  (TILE_H + 2)         // 18
#define LDS_W   36                   // padded LDS stride (floats)
#define HALO_ELEMS (HALO_W * HALO_H) // 612

typedef float v4f __attribute__((ext_vector_type(4)));
typedef __attribute__((address_space(3))) float lds_float_t;

__global__ __launch_bounds__(256)
void upfirdn2d_up2_k4_kernel(const float* __restrict__ x,
                             const float* __restrict__ kern,
                             float* __restrict__ out)
{
    __shared__ float tile[HALO_H * LDS_W];   // 18*36*4 = 2592 B

    const int tid   = threadIdx.x;
    const int plane = blockIdx.y;                    // b*C + c  (0..511)
    const int tiles_x = W_IN / TILE_W;               // 8
    const int tx0 = (blockIdx.x % tiles_x) * TILE_W;
    const int ty0 = (blockIdx.x / tiles_x) * TILE_H;

    const float* __restrict__ xp = x   + (size_t)plane * H_IN * W_IN;
    float*       __restrict__ op = out + (size_t)plane * W_OUT * W_OUT;

    // ---- Stage (TILE + 1-halo) input region into LDS with CDNA5 async copies ----
    #pragma unroll
    for (int it = 0; it < 3; ++it) {
        int idx = tid + it * 256;
        if (idx < HALO_ELEMS) {
            const int yr = idx / HALO_W;
            const int xr = idx - yr * HALO_W;
            const int gy = ty0 - 1 + yr;
            const int gx = tx0 - 1 + xr;
            const int loff = yr * LDS_W + xr;
            if ((unsigned)gy < (unsigned)H_IN && (unsigned)gx < (unsigned)W_IN) {
                const float* gp = xp + gy * W_IN + gx;
                // LDS byte offset of destination (addrspace(3) strip -> 32-bit offset)
                unsigned ldsoff = (unsigned)(size_t)(lds_float_t*)(&tile[loff]);
                // GV form: 64-bit global address in VADDR, LDS byte offset in VDST (ASYNCcnt)
                asm volatile("global_load_async_to_lds_b32 %0, %1, off"
                             :: "v"(ldsoff), "v"(gp)
                             : "memory");
            } else {
                tile[loff] = 0.0f;   // zero halo outside the tensor
            }
        }
    }
    asm volatile("s_wait_asynccnt 0" ::: "memory");   // this wave's async copies done
    __syncthreads();                                   // whole tile visible

    // ---- FIR weights (uniform address -> scalarized loads) ----
    const float k00=kern[0],  k01=kern[1],  k02=kern[2],  k03=kern[3];
    const float k10=kern[4],  k11=kern[5],  k12=kern[6],  k13=kern[7];
    const float k20=kern[8],  k21=kern[9],  k22=kern[10], k23=kern[11];
    const float k30=kern[12], k31=kern[13], k32=kern[14], k33=kern[15];

    // ---- Each thread: 3x4 LDS neighborhood -> two adjacent 2x2 output quads ----
    const int ltx = tid & 15;        // input col pair index (cols 2*ltx, 2*ltx+1)
    const int lty = tid >> 4;        // input row within tile (0..15)
    const int lb  = lty * LDS_W + 2 * ltx;   // LDS addr of x[iy-1, ix-1]

    const float b00 = tile[lb + 0],         b01 = tile[lb + 1];
    const float b02 = tile[lb + 2],         b03 = tile[lb + 3];
    const float b10 = tile[lb + LDS_W],     b11 = tile[lb + LDS_W + 1];
    const float b12 = tile[lb + LDS_W + 2], b13 = tile[lb + LDS_W + 3];
    const float b20 = tile[lb + 2*LDS_W],   b21 = tile[lb + 2*LDS_W + 1];
    const float b22 = tile[lb + 2*LDS_W+2], b23 = tile[lb + 2*LDS_W + 3];

    // out[2iy+r, 2ix+s] = sum_{dy,dx} k[r+2dy][s+2dx] * x[iy-1+r+dy, ix-1+s+dx]
    // quad 0: input pixel (iy, ix)
    const float o00 = k00*b00 + k02*b01 + k20*b10 + k22*b11;
    const float o01 = k01*b01 + k03*b02 + k21*b11 + k23*b12;
    const float o10 = k10*b10 + k12*b11 + k30*b20 + k32*b21;
    const float o11 = k11*b11 + k13*b12 + k31*b21 + k33*b22;
    // quad 1: input pixel (iy, ix+1)  (shift columns by one)
    const float p00 = k00*b01 + k02*b02 + k20*b11 + k22*b12;
    const float p01 = k01*b02 + k03*b03 + k21*b12 + k23*b13;
    const float p10 = k10*b11 + k12*b12 + k30*b21 + k32*b22;
    const float p11 = k11*b12 + k13*b13 + k31*b22 + k33*b23;

    const int oy = 2 * (ty0 + lty);
    const int ox = 2 * (tx0 + 2 * ltx);     // multiple of 4 -> 16B aligned

    v4f r0; r0.x = o00; r0.y = o01; r0.z = p00; r0.w = p01;
    v4f r1; r1.x = o10; r1.y = o11; r1.z = p10; r1.w = p11;
    // Streaming write-once output: non-temporal, coalesced 128-bit stores
    __builtin_nontemporal_store(r0, (v4f*)(op + (size_t)oy       * W_OUT + ox));
    __builtin_nontemporal_store(r1, (v4f*)(op + (size_t)(oy + 1) * W_OUT + ox));
}

extern "C" void kernel_launch(void* const* d_in, const int* in_sizes, int n_in,
                              void* d_out, int out_size, void* d_ws, size_t ws_size,
                              hipStream_t stream) {
    const float* x    = (const float*)d_in[0];   // (8,64,256,256)
    const float* kern = (const float*)d_in[1];   // (4,4)
    float*       out  = (float*)d_out;           // (8,64,512,512)

    const int planes  = 8 * 64;                                  // 512
    const int tiles   = (H_IN / TILE_H) * (W_IN / TILE_W);       // 16*8 = 128

    dim3 grid(tiles, planes, 1);
    dim3 block(256, 1, 1);
    upfirdn2d_up2_k4_kernel<<<grid, block, 0, stream>>>(x, kern, out);
}